// ListMultiScaleRoIAlign_1168231105001
// MI455X (gfx1250) — compile-verified
//
#include <hip/hip_runtime.h>
#include <cstdint>

typedef float v2f __attribute__((ext_vector_type(2)));
typedef float v8f __attribute__((ext_vector_type(8)));

#define CTILE 16

struct BoxInfo {
    float x1s, y1s, bw, bh;
    int   idx, b, Wd, pad;
};

// ---------------------------------------------------------------------------
// Kernel 1: per-box FPN level assignment + scaled geometry (one thread/box).
// ---------------------------------------------------------------------------
__global__ void prep_boxes(const float* __restrict__ boxes, BoxInfo* __restrict__ bi,
                           int nbox, int Kbox) {
    int k = blockIdx.x * blockDim.x + threadIdx.x;
    if (k >= nbox) return;
    float x1 = boxes[4*k+0], y1 = boxes[4*k+1], x2 = boxes[4*k+2], y2 = boxes[4*k+3];
    float area = (x2 - x1) * (y2 - y1);
    float lvl  = floorf(4.0f + log2f(sqrtf(area) / 224.0f + 1e-6f));
    lvl = fminf(fmaxf(lvl, 2.0f), 5.0f);
    int idx = (int)lvl - 2;
    float scale = 0.25f / (float)(1 << idx);
    BoxInfo o;
    o.x1s = x1 * scale;
    o.y1s = y1 * scale;
    float rw = fmaxf(x2 * scale - o.x1s, 1.0f);
    float rh = fmaxf(y2 * scale - o.y1s, 1.0f);
    o.bw  = rw * (1.0f / 7.0f);
    o.bh  = rh * (1.0f / 7.0f);
    o.idx = idx;
    o.b   = k / Kbox;
    o.Wd  = 200 >> idx;        // 200,100,50,25 (square levels)
    o.pad = 0;
    bi[k] = o;
}

// ---------------------------------------------------------------------------
// Kernel 2: one wave32 per (box, 16-channel tile).
//   out_c(7x7) = YW(7xHr) * F_c(HrxWr) * XW(Wrx7)   via V_WMMA_F32_16X16X4_F32
// XW/YW carry bilinear taps * 0.5 pooling * validity mask.
// ---------------------------------------------------------------------------
__global__ __launch_bounds__(32) void roi_align_wmma(
    const float* __restrict__ f0, const float* __restrict__ f1,
    const float* __restrict__ f2, const float* __restrict__ f3,
    const BoxInfo* __restrict__ binfo, float* __restrict__ out) {

    __shared__ float XW[64 * 16];   // [col][n]  (n = pw, rows 7..15 stay zero)
    __shared__ float YW[16 * 64];   // [n][row]  (n = ph, rows 7..15 stay zero)
    __shared__ float U [4 * 16 * 8];// GEMM1->GEMM2 bounce: [k][c][pw]
    __shared__ float OB[CTILE * 49];// (c, ph, pw) staging for coalesced store

    const int lane = threadIdx.x;
    const int box  = blockIdx.x >> 4;
    const int c0   = (blockIdx.x & 15) * CTILE;

    const BoxInfo bi = binfo[box];
    const float* feats[4] = {f0, f1, f2, f3};
    const int Wd = bi.Wd, Hd = bi.Wd;
    const size_t plane = (size_t)Wd * (size_t)Hd;
    const float* F = feats[bi.idx] + (size_t)(bi.b * 256 + c0) * plane;

    for (int i = lane; i < 64 * 16; i += 32) { XW[i] = 0.0f; YW[i] = 0.0f; }
    __syncthreads();

    // Touched row/col ranges (wave-uniform). Samples are monotone in j.
    float xsA = bi.x1s + 0.25f * bi.bw, xsB = bi.x1s + 6.75f * bi.bw;
    int xlo = (int)floorf(fminf(fmaxf(xsA, 0.0f), (float)(Wd - 1)));
    int xhi = min((int)floorf(fminf(fmaxf(xsB, 0.0f), (float)(Wd - 1))) + 1, Wd - 1);
    int Kx  = min(((xhi - xlo + 1) + 3) & ~3, 64);
    float ysA = bi.y1s + 0.25f * bi.bh, ysB = bi.y1s + 6.75f * bi.bh;
    int ylo = (int)floorf(fminf(fmaxf(ysA, 0.0f), (float)(Hd - 1)));
    int yhi = min((int)floorf(fminf(fmaxf(ysB, 0.0f), (float)(Hd - 1))) + 1, Hd - 1);
    int Ky  = min(((yhi - ylo + 1) + 3) & ~3, 64);

    // Build XW (lanes 0..13: 14 x-samples) and YW (lanes 16..29: 14 y-samples).
    {
        int j = lane & 15;
        if (lane < 14) {
            float xs = bi.x1s + ((float)j + 0.5f) * 0.5f * bi.bw;
            float v  = (xs >= -1.0f && xs <= (float)Wd) ? 0.5f : 0.0f;
            float x  = fminf(fmaxf(xs, 0.0f), (float)(Wd - 1));
            int x0i  = (int)floorf(x);
            int x1i  = min(x0i + 1, Wd - 1);
            float lx = x - (float)x0i;
            int pw = j >> 1;
            int cA = x0i - xlo, cB = x1i - xlo;
            if (cA >= 0 && cA < 64) atomicAdd(&XW[cA * 16 + pw], v * (1.0f - lx));
            if (cB >= 0 && cB < 64) atomicAdd(&XW[cB * 16 + pw], v * lx);
        } else if (lane >= 16 && j < 14) {
            float ys = bi.y1s + ((float)j + 0.5f) * 0.5f * bi.bh;
            float v  = (ys >= -1.0f && ys <= (float)Hd) ? 0.5f : 0.0f;
            float y  = fminf(fmaxf(ys, 0.0f), (float)(Hd - 1));
            int y0i  = (int)floorf(y);
            int y1i  = min(y0i + 1, Hd - 1);
            float ly = y - (float)y0i;
            int ph = j >> 1;
            int rA = y0i - ylo, rB = y1i - ylo;
            if (rA >= 0 && rA < 64) atomicAdd(&YW[ph * 64 + rA], v * (1.0f - ly));
            if (rB >= 0 && rB < 64) atomicAdd(&YW[ph * 64 + rB], v * ly);
        }
    }
    __syncthreads();

    // WMMA lane roles (16x16x4 f32): lanes 0-15 hold K{0,1}, lanes 16-31 K{2,3}.
    const int n  = lane & 15;   // M row (channel) for A, N col for B/D
    const int hi = lane >> 4;   // K-half selector
    const float* Fc = F + (size_t)n * plane;

    v8f O[7] = {};              // GEMM2 accumulators, one per pw

    const int NXC = Kx >> 2, NYC = Ky >> 2;
    for (int yc = 0; yc < NYC; ++yc) {
        v8f Dy[4];
        for (int ky = 0; ky < 4; ++ky) {
            int y = min(ylo + yc * 4 + ky, Hd - 1);
            const float* rowp = Fc + (size_t)y * Wd;
            __builtin_prefetch(rowp + Wd, 0, 1);       // global_prefetch_b8: next row
            v8f d = {};
            for (int xc = 0; xc < NXC; ++xc) {
                int xb = min(xlo + xc * 4 + hi * 2, Wd - 2);
                v2f a; a.x = rowp[xb]; a.y = rowp[xb + 1];
                int col0 = xc * 4 + hi * 2;
                v2f b; b.x = XW[col0 * 16 + n]; b.y = XW[(col0 + 1) * 16 + n];
                d = __builtin_amdgcn_wmma_f32_16x16x4_f32(
                        false, a, false, b, (short)0, d, false, false);
            }
            Dy[ky] = d;
        }
        // Bounce D-layout (M=r+8*hi, N=lane&15) into U[k][c][pw] for A-reads.
        if (n < 7) {
            for (int ky = 0; ky < 4; ++ky)
                for (int r = 0; r < 8; ++r)
                    U[(ky * 16 + (r + 8 * hi)) * 8 + n] = Dy[ky][r];
        }
        __syncthreads();
        v2f by;
        by.x = YW[n * 64 + yc * 4 + 2 * hi + 0];
        by.y = YW[n * 64 + yc * 4 + 2 * hi + 1];
        for (int pw = 0; pw < 7; ++pw) {
            v2f a;
            a.x = U[((2 * hi + 0) * 16 + n) * 8 + pw];
            a.y = U[((2 * hi + 1) * 16 + n) * 8 + pw];
            O[pw] = __builtin_amdgcn_wmma_f32_16x16x4_f32(
                        false, a, false, by, (short)0, O[pw], false, false);
        }
        __syncthreads();
    }

    // D of GEMM2: M = channel = r + 8*hi, N = ph = lane&15. Stage then store.
    if (n < 7) {
        for (int pw = 0; pw < 7; ++pw)
            for (int r = 0; r < 8; ++r)
                OB[(r + 8 * hi) * 49 + n * 7 + pw] = O[pw][r];
    }
    __syncthreads();
    float* op = out + ((size_t)box * 256 + c0) * 49;
    for (int t = lane; t < CTILE * 49; t += 32) op[t] = OB[t];
}

// ---------------------------------------------------------------------------
extern "C" void kernel_launch(void* const* d_in, const int* in_sizes, int n_in,
                              void* d_out, int out_size, void* d_ws, size_t ws_size,
                              hipStream_t stream) {
    const float* f0    = (const float*)d_in[0];
    const float* f1    = (const float*)d_in[1];
    const float* f2    = (const float*)d_in[2];
    const float* f3    = (const float*)d_in[3];
    const float* boxes = (const float*)d_in[4];

    int B = in_sizes[0] / (256 * 200 * 200);
    if (B < 1) B = 1;
    int nbox = in_sizes[4] / 4;
    int Kbox = nbox / B;

    BoxInfo* bi = (BoxInfo*)d_ws;
    prep_boxes<<<(nbox + 255) / 256, 256, 0, stream>>>(boxes, bi, nbox, Kbox);
    roi_align_wmma<<<dim3((unsigned)nbox * 16), dim3(32), 0, stream>>>(
        f0, f1, f2, f3, bi, (float*)d_out);
}